// GNNNodeClassifierImproved_77154792506052
// MI455X (gfx1250) — compile-verified
//
#include <hip/hip_runtime.h>

#define NN      50000
#define FIN     128
#define HID     96
#define HEADS   4
#define CHEAD   24
#define CLS     8
#define NEG_SLOPE 0.2f
#define BN_EPS  1e-5f

typedef __attribute__((ext_vector_type(2))) float v2f;
typedef __attribute__((ext_vector_type(8))) float v8f;
typedef unsigned int u32x4 __attribute__((ext_vector_type(4)));
typedef int          i32x4 __attribute__((ext_vector_type(4)));
typedef int          i32x8 __attribute__((ext_vector_type(8)));

// ---------------------------------------------------------------------------
// GEMM: H[M,96] = X[M,K] @ W[K,96] using V_WMMA_F32_16X16X4_F32.
// One block = 192 threads = 6 waves; block handles a 16-row tile of nodes,
// wave w handles output columns [16w, 16w+16).
// A tile is DMA'd into LDS by the Tensor Data Mover (one tensor_load_to_lds
// issued by wave 0, tracked with TENSORcnt). TDM pad_enable inserts 4 dwords
// of padding so fragment reads are LDS-bank-conflict-free.
//   K=128: pad every 128 dwords (row stride 132) -> banks (4r+k)%64 distinct.
//   K=96 : pad every 32 dwords  (addr 108r+k+4*(k/32)) -> 44r%64 distinct.
// ---------------------------------------------------------------------------
template <int K, int LOGI>
__device__ __forceinline__ int lds_idx(int r, int k) {
  int lin = r * K + k;
  return lin + ((lin >> LOGI) << 2);   // + 4 dwords per pad interval
}

template <int K, int LOGI>
__global__ __launch_bounds__(192)
void gemm_wmma_kernel(const float* __restrict__ X, const float* __restrict__ W,
                      float* __restrict__ H, int nrows) {
  constexpr int PADTOT = ((16 * K) >> LOGI) * 4;
  __shared__ float As[16 * K + PADTOT + 8];

  const int row0 = blockIdx.x * 16;

  if (threadIdx.x < 32) {
    // ---- build Tensor DMA descriptor (D#) for a 16 x K f32 tile ----
    unsigned long gaddr = (unsigned long)(X + (long)row0 * K);
    u32x4 g0;
    g0[0] = 1u;                                      // count=1, user mode
    g0[1] = 0u;                                      // lds_addr = 0 (As base)
    g0[2] = (unsigned)(gaddr & 0xffffffffu);         // global_addr[31:0]
    g0[3] = (unsigned)((gaddr >> 32) & 0x01ffffffu)  // global_addr[56:32]
            | (2u << 30);                            // type = 2 (image)
    unsigned dim0 = (unsigned)K;                     // tensor dim0 (elems)
    unsigned dim1 = (unsigned)(nrows - row0);        // remaining rows
    i32x8 g1;
    g1[0] = (int)((2u << 16)                         // data_size = 4 bytes
                  | (1u << 20)                       // pad_enable
                  | ((unsigned)(LOGI - 1) << 22)     // pad_interval code
                  | (3u << 25));                     // pad_amount = 4 dwords
    g1[1] = (int)((dim0 & 0xffffu) << 16);           // tensor_dim0[15:0]
    g1[2] = (int)((dim0 >> 16) | ((dim1 & 0xffffu) << 16)); // dim0 hi | dim1 lo
    g1[3] = (int)((dim1 >> 16) | ((unsigned)K << 16));      // dim1 hi | tile_dim0=K
    g1[4] = 16;                                      // tile_dim1=16, tile_dim2=0
    g1[5] = (int)K;                                  // tensor_dim0_stride lo32
    g1[6] = 0;                                       // stride hi | dim1_stride lo
    g1[7] = 0;                                       // dim1_stride hi (unused)
    i32x4 g2 = {0, 0, 0, 0};                         // 3rd/4th dims unused
    i32x4 g3 = {0, 0, 0, 0};
    i32x8 g4 = {0, 0, 0, 0, 0, 0, 0, 0};             // extra group (unused)
    __builtin_amdgcn_tensor_load_to_lds(g0, g1, g2, g3, g4, 0);
    __builtin_amdgcn_s_wait_tensorcnt(0);
  }
  __syncthreads();

  const int wave = threadIdx.x >> 5;       // 0..5 -> N tile
  const int lane = threadIdx.x & 31;
  const int r16  = lane & 15;              // row (A) / col (B) within tile
  const int half = lane >> 4;              // 0: K pair {0,1}, 1: K pair {2,3}
  const int colg = wave * 16 + r16;        // global output column

  v8f acc = {};
  #pragma unroll
  for (int kk = 0; kk < K; kk += 4) {
    const int k0 = kk + 2 * half;
    v2f a, b;
    // A 16x4 layout: v0 = {K=0 | K=2}, v1 = {K=1 | K=3}
    a.x = As[lds_idx<K, LOGI>(r16, k0)];
    a.y = As[lds_idx<K, LOGI>(r16, k0 + 1)];
    // B 4x16 layout (row striped across lanes, symmetric K split)
    b.x = W[(long)k0 * HID + colg];
    b.y = W[(long)(k0 + 1) * HID + colg];
    acc = __builtin_amdgcn_wmma_f32_16x16x4_f32(
        /*neg_a=*/false, a, /*neg_b=*/false, b,
        /*c_mod=*/(short)0, acc, /*reuse_a=*/false, /*reuse_b=*/false);
  }

  // D layout: VGPR r holds M=r (lanes 0-15) / M=r+8 (lanes 16-31), N=lane%16
  const int rbase = row0 + 8 * half;
  #pragma unroll
  for (int r = 0; r < 8; ++r) {
    H[(long)(rbase + r) * HID + colg] = acc[r];
  }
}

// ---------------------------------------------------------------------------
// alpha_s/alpha_d: per (node, head) dot of h[n,h,:] with a_src/a_dst
// ---------------------------------------------------------------------------
__global__ void alpha_kernel(const float* __restrict__ h,
                             const float* __restrict__ a_src,
                             const float* __restrict__ a_dst,
                             float* __restrict__ alpha_s,
                             float* __restrict__ alpha_d, int n) {
  int t = blockIdx.x * blockDim.x + threadIdx.x;
  if (t >= n * HEADS) return;
  int node = t >> 2, hd = t & 3;
  const float* hp = h + (long)node * HID + hd * CHEAD;
  const float* sp = a_src + hd * CHEAD;
  const float* dp = a_dst + hd * CHEAD;
  float s = 0.f, d = 0.f;
  #pragma unroll
  for (int c = 0; c < CHEAD; ++c) { s += hp[c] * sp[c]; d += hp[c] * dp[c]; }
  alpha_s[t] = s;
  alpha_d[t] = d;
}

// order-preserving uint encoding for deterministic float atomic-max
__device__ __forceinline__ unsigned ordkey(float f) {
  unsigned u = __float_as_uint(f);
  return (u & 0x80000000u) ? ~u : (u | 0x80000000u);
}
__device__ __forceinline__ float ordval(unsigned k) {
  unsigned u = (k & 0x80000000u) ? (k ^ 0x80000000u) : ~k;
  return __uint_as_float(u);
}

__device__ __forceinline__ void edge_pair(const int* __restrict__ ei, int e, int E,
                                          int& src, int& dst) {
  if (e < E) { src = ei[e]; dst = ei[E + e]; }   // edge_index row-major [2,E]
  else       { src = dst = e - E; }              // self-loop
}

// pass A: m[dst,h] = max over incoming edges of leaky_relu(as[src]+ad[dst])
__global__ void edge_max_kernel(const int* __restrict__ ei,
                                const float* __restrict__ as_,
                                const float* __restrict__ ad_,
                                unsigned* __restrict__ mkey, int E, int n) {
  int t = blockIdx.x * blockDim.x + threadIdx.x;
  if (t >= (E + n) * HEADS) return;
  int e = t >> 2, hd = t & 3;
  int src, dst; edge_pair(ei, e, E, src, dst);
  float v = as_[src * HEADS + hd] + ad_[dst * HEADS + hd];
  v = v > 0.f ? v : NEG_SLOPE * v;
  atomicMax(&mkey[dst * HEADS + hd], ordkey(v));
}

// pass B: agg[dst,ch] += exp(e-m)*h[src,ch];  den[dst,h] += exp(e-m)
__global__ void edge_aggregate_kernel(const int* __restrict__ ei,
                                      const float* __restrict__ as_,
                                      const float* __restrict__ ad_,
                                      const unsigned* __restrict__ mkey,
                                      const float* __restrict__ h,
                                      float* __restrict__ agg,
                                      float* __restrict__ den, int E, int n) {
  long t = (long)blockIdx.x * blockDim.x + threadIdx.x;
  long total = (long)(E + n) * HID;
  if (t >= total) return;
  int ch = (int)(t % HID);
  int e  = (int)(t / HID);
  int hd = ch / CHEAD;
  int src, dst; edge_pair(ei, e, E, src, dst);
  float v = as_[src * HEADS + hd] + ad_[dst * HEADS + hd];
  v = v > 0.f ? v : NEG_SLOPE * v;
  float m  = ordval(mkey[dst * HEADS + hd]);
  float ex = __expf(v - m);
  atomicAdd(&agg[(long)dst * HID + ch], ex * h[(long)src * HID + ch]);
  if ((ch % CHEAD) == 0) atomicAdd(&den[dst * HEADS + hd], ex);
}

// normalize (in place) + bias
__global__ void normalize_kernel(float* __restrict__ agg,
                                 const float* __restrict__ den,
                                 const float* __restrict__ bias, int n) {
  long t = (long)blockIdx.x * blockDim.x + threadIdx.x;
  if (t >= (long)n * HID) return;
  int ch = (int)(t % HID);
  int node = (int)(t / HID);
  int hd = ch / CHEAD;
  agg[t] = agg[t] / (den[node * HEADS + hd] + 1e-16f) + bias[ch];
}

// BN stats: per-block partial sums (one channel per thread, blockDim = 96)
#define STAT_ROWS 64
__global__ __launch_bounds__(HID)
void bn_stats_kernel(const float* __restrict__ x, float* __restrict__ sums,
                     float* __restrict__ sumsq, int n) {
  int ch = threadIdx.x;
  int r0 = blockIdx.x * STAT_ROWS;
  int r1 = r0 + STAT_ROWS; if (r1 > n) r1 = n;
  float s = 0.f, q = 0.f;
  for (int r = r0; r < r1; ++r) {
    float v = x[(long)r * HID + ch];
    s += v; q += v * v;
  }
  atomicAdd(&sums[ch], s);
  atomicAdd(&sumsq[ch], q);
}

__global__ void bn_finalize_kernel(float* __restrict__ sums,
                                   float* __restrict__ sumsq, int n) {
  int ch = threadIdx.x;
  if (ch >= HID) return;
  float mu  = sums[ch] / (float)n;
  float var = sumsq[ch] / (float)n - mu * mu;
  sums[ch]  = mu;                       // reuse: sums <- mu
  sumsq[ch] = rsqrtf(var + BN_EPS);     // reuse: sumsq <- 1/sigma
}

__global__ void bn_apply_elu_kernel(const float* __restrict__ x,
                                    const float* __restrict__ mu,
                                    const float* __restrict__ inv,
                                    const float* __restrict__ gamma,
                                    const float* __restrict__ beta,
                                    float* __restrict__ y, int n) {
  long t = (long)blockIdx.x * blockDim.x + threadIdx.x;
  if (t >= (long)n * HID) return;
  int ch = (int)(t % HID);
  float v = gamma[ch] * (x[t] - mu[ch]) * inv[ch] + beta[ch];
  y[t] = v > 0.f ? v : (__expf(v) - 1.0f);
}

__global__ void fc_kernel(const float* __restrict__ h,
                          const float* __restrict__ W,
                          const float* __restrict__ b,
                          float* __restrict__ out, int n) {
  int t = blockIdx.x * blockDim.x + threadIdx.x;
  if (t >= n * CLS) return;
  int node = t / CLS, cls = t % CLS;
  float s = b[cls];
  const float* hp = h + (long)node * HID;
  #pragma unroll 8
  for (int c = 0; c < HID; ++c) s += hp[c] * W[c * CLS + cls];
  out[t] = s;
}

__global__ void fill_zero_kernel(float* __restrict__ p, long nel) {
  long t = (long)blockIdx.x * blockDim.x + threadIdx.x;
  if (t < nel) p[t] = 0.f;
}

// ---------------------------------------------------------------------------
static inline int cdiv(long a, int b) { return (int)((a + b - 1) / b); }

extern "C" void kernel_launch(void* const* d_in, const int* in_sizes, int n_in,
                              void* d_out, int out_size, void* d_ws, size_t ws_size,
                              hipStream_t stream) {
  const int n = in_sizes[0] / FIN;     // 50000
  const int E = in_sizes[1] / 2;       // 800000
  const float* x  = (const float*)d_in[0];
  const int*   ei = (const int*)d_in[1];
  const float* fcW = (const float*)d_in[20];
  const float* fcb = (const float*)d_in[21];

  // workspace carve-up (floats)
  float* ws = (float*)d_ws;
  const long NF = (long)n * HID;
  float*    buf0    = ws;                       // layer input (layers 1,2) / bn output
  float*    hlin    = ws + NF;                  // gemm output h
  float*    agg     = ws + 2 * NF;              // aggregation / post buffer
  float*    alpha_s = ws + 3 * NF;
  float*    alpha_d = alpha_s + (long)n * HEADS;
  unsigned* mkey    = (unsigned*)(alpha_d + (long)n * HEADS);
  float*    den     = (float*)(mkey + (long)n * HEADS);
  float*    sums    = den + (long)n * HEADS;    // 96
  float*    sumsq   = sums + HID;               // 96
  (void)ws_size; (void)n_in;

  const float* in = x;
  for (int L = 0; L < 3; ++L) {
    const float* W  = (const float*)d_in[2 + 6 * L];
    const float* as = (const float*)d_in[3 + 6 * L];
    const float* ad = (const float*)d_in[4 + 6 * L];
    const float* bi = (const float*)d_in[5 + 6 * L];
    const float* g  = (const float*)d_in[6 + 6 * L];
    const float* bt = (const float*)d_in[7 + 6 * L];

    // 1. h = in @ W  (TDM tile load + WMMA f32)
    if (L == 0)
      gemm_wmma_kernel<FIN, 7><<<n / 16, 192, 0, stream>>>(in, W, hlin, n);
    else
      gemm_wmma_kernel<HID, 5><<<n / 16, 192, 0, stream>>>(in, W, hlin, n);

    // 2. per-node attention scalars
    alpha_kernel<<<cdiv((long)n * HEADS, 256), 256, 0, stream>>>(
        hlin, as, ad, alpha_s, alpha_d, n);

    // 3. zero accumulators: agg (NF) and [mkey|den|sums|sumsq] (contiguous)
    fill_zero_kernel<<<cdiv(NF, 256), 256, 0, stream>>>(agg, NF);
    fill_zero_kernel<<<cdiv((long)n * HEADS * 2 + 2 * HID, 256), 256, 0, stream>>>(
        (float*)mkey, (long)n * HEADS * 2 + 2 * HID);

    // 4. segment max (per dst, per head)
    edge_max_kernel<<<cdiv((long)(E + n) * HEADS, 256), 256, 0, stream>>>(
        ei, alpha_s, alpha_d, mkey, E, n);

    // 5. unnormalized aggregation + denominator
    edge_aggregate_kernel<<<cdiv((long)(E + n) * HID, 256), 256, 0, stream>>>(
        ei, alpha_s, alpha_d, mkey, hlin, agg, den, E, n);

    // 6. normalize + bias (in place on agg)
    normalize_kernel<<<cdiv(NF, 256), 256, 0, stream>>>(agg, den, bi, n);

    // 7. batch-norm stats -> finalize -> apply + ELU into buf0
    bn_stats_kernel<<<cdiv(n, STAT_ROWS), HID, 0, stream>>>(agg, sums, sumsq, n);
    bn_finalize_kernel<<<1, HID, 0, stream>>>(sums, sumsq, n);
    bn_apply_elu_kernel<<<cdiv(NF, 256), 256, 0, stream>>>(
        agg, sums, sumsq, g, bt, buf0, n);

    in = buf0;
  }

  // final classifier
  fc_kernel<<<cdiv((long)n * CLS, 256), 256, 0, stream>>>(
      buf0, fcW, fcb, (float*)d_out, n);
  (void)out_size;
}